// Test_EinsumSlice_62878321214346
// MI455X (gfx1250) — compile-verified
//
#include <hip/hip_runtime.h>

typedef __attribute__((ext_vector_type(2))) float v2f;
typedef __attribute__((ext_vector_type(8))) float v8f;

#define KDIM   4096   // H
#define NOUT   8192   // 2*INTER
#define KC     8      // deterministic k-split
#define KCHUNK (KDIM / KC)        // 512
#define WAVES_PER_WG 8
#define TILE_N 16

// out_partial[kc][n] = sum over k in chunk kc of x[k] * w[k*NOUT + n]
// Implemented as degenerate 16x16x4 fp32 WMMA: only row M=0 of A is nonzero.
__global__ __launch_bounds__(256)
void Test_EinsumSlice_gemv_wmma(const float* __restrict__ x,
                                const float* __restrict__ w,
                                float* __restrict__ partial) {
    const int lane = threadIdx.x & 31;
    const int wave = threadIdx.x >> 5;
    const int l16  = lane & 15;
    const int hi   = lane >> 4;          // 0: K+0/K+1 rows, 1: K+2/K+3 rows

    const int n0 = (blockIdx.x * WAVES_PER_WG + wave) * TILE_N;
    const int k0 = blockIdx.y * KCHUNK;

    // A-matrix (16x4 f32): lanes 0-15 hold M=0..15 {K0,K1}; lanes 16-31 {K2,K3}.
    // Only M=0 (l16==0) carries x; mask the rest to zero.
    const float msk = (l16 == 0) ? 1.0f : 0.0f;

    // B-matrix (4x16 f32): VGPR0 = rows K(+2*hi), VGPR1 = rows K+1(+2*hi), N = l16.
    const float* wp = w + (size_t)(k0 + 2 * hi) * NOUT + (n0 + l16);
    const float* xp = x + (k0 + 2 * hi);

    v8f acc[4];
    #pragma unroll
    for (int u = 0; u < 4; ++u) acc[u] = (v8f){0.f,0.f,0.f,0.f,0.f,0.f,0.f,0.f};

    for (int kk = 0; kk < KCHUNK; kk += 16) {
        #pragma unroll
        for (int u = 0; u < 4; ++u) {
            v2f A, B;
            A[0] = xp[4 * u + 0] * msk;
            A[1] = xp[4 * u + 1] * msk;
            B[0] = wp[(4 * u + 0) * NOUT];
            B[1] = wp[(4 * u + 1) * NOUT];
            acc[u] = __builtin_amdgcn_wmma_f32_16x16x4_f32(
                false, A, false, B, (short)0, acc[u], false, false);
        }
        wp += 16 * NOUT;
        xp += 16;
    }

    v8f c = (acc[0] + acc[1]) + (acc[2] + acc[3]);

    // D layout: VGPR0, lanes 0-15 = row M=0, N=lane.
    if (lane < 16) {
        partial[(size_t)blockIdx.y * NOUT + n0 + lane] = c[0];
    }
}

__global__ __launch_bounds__(256)
void Test_EinsumSlice_reduce(const float* __restrict__ partial,
                             float* __restrict__ out) {
    const int n = blockIdx.x * blockDim.x + threadIdx.x;
    if (n < NOUT) {
        float s = 0.0f;
        #pragma unroll
        for (int c = 0; c < KC; ++c) s += partial[(size_t)c * NOUT + n];
        out[n] = s;
    }
}

extern "C" void kernel_launch(void* const* d_in, const int* in_sizes, int n_in,
                              void* d_out, int out_size, void* d_ws, size_t ws_size,
                              hipStream_t stream) {
    // Inputs (reference setup_inputs order):
    //   d_in[0] = x, fp32, shape (1,1,2048,4096) -> we use row m=0: x[0..4095]
    //   d_in[1] = w, fp32, shape (1,8,4096,8192) -> expert 0 slice at offset 0
    const float* x = (const float*)d_in[0];
    const float* w = (const float*)d_in[1];
    float* out = (float*)d_out;
    float* partial = (float*)d_ws;   // KC * NOUT * 4 = 256 KB

    dim3 grid1(NOUT / (TILE_N * WAVES_PER_WG), KC);   // (64, 8)
    Test_EinsumSlice_gemv_wmma<<<grid1, 256, 0, stream>>>(x, w, partial);

    dim3 grid2((NOUT + 255) / 256);                   // 32
    Test_EinsumSlice_reduce<<<grid2, 256, 0, stream>>>(partial, out);
}